// SpectralConv1d_5961414607122
// MI455X (gfx1250) — compile-verified
//
#include <hip/hip_runtime.h>
#include <math.h>

// ---------------------------------------------------------------------------
// SpectralConv1d (FNO layer) for MI455X / gfx1250.
//
// B=32, Cin=Cout=64, L=8192, modes=25 (k=0..12 and k=L-12..L-1).
//   X = x * Wb          (2048 x 8192) * (8192 x 32)  f32 WMMA, K-split+atomics
//   G = mix(X, weights)  tiny complex per-mode matmul (VALU)
//   y = G * Wt          (2048 x 32)  * (32 x 8192)   f32 WMMA, store-bound
// Basis columns j=0..12 : cos(2*pi*j*t/L), j=16..28 : sin(2*pi*(j-16)*t/L), rest 0.
// Basis tables are stored FRAGMENT-PACKED so every B fragment is one
// contiguous b64 load per lane (wave-contiguous 128B/256B).
// ---------------------------------------------------------------------------

typedef __attribute__((ext_vector_type(2))) float v2f;
typedef __attribute__((ext_vector_type(8))) float v8f;

#define L_LEN   8192
#define NB      32
#define NC      64
#define NMODES  25
#define JCOLS   32          // padded basis columns (13 cos + pad + 13 sin + pad)
#define NROWS   (NB * NC)   // 2048 rows = (b, c) or (b, o)
#define KCHUNKS 32          // stage-1 K split (K=256 per chunk)

// basis value: j<=12 -> cos(2*pi*j*t/L); 16<=j<=28 -> sin(2*pi*(j-16)*t/L); else 0
__device__ __forceinline__ float basis_val(int j, int t) {
    const float w0 = 6.28318530717958647692f / (float)L_LEN;
    if (j <= 12) {
        const int r = (j * t) & (L_LEN - 1);          // arg in [0, 2*pi)
        return __builtin_cosf(w0 * (float)r);
    }
    if (j >= 16 && j <= 28) {
        const int r = ((j - 16) * t) & (L_LEN - 1);
        return __builtin_sinf(w0 * (float)r);
    }
    return 0.0f;
}

// ---------------------------------------------------------------------------
// Fragment-packed basis tables (one thread per t = 0..8191):
//  Analysis  Wbf: B fragment for K-group g (K=4g..4g+3), lane=(half,j):
//     flat[(g*2 + half)*64 + j*2 + c] = basis(j, t=4g+2*half+c)
//  Synthesis Wtf: B fragment for col tile ct, K-group kg (j=4kg..4kg+3):
//     flat[ct*512 + kg*64 + half*32 + l*2 + c] = basis(4kg+2*half+c, t=ct*16+l)
// Each table: 262144 floats (1 MB).
// ---------------------------------------------------------------------------
__global__ void gen_basis(float* __restrict__ Wbf, float* __restrict__ Wtf) {
    const int t = blockIdx.x * blockDim.x + threadIdx.x;
    if (t >= L_LEN) return;
    // analysis table: t -> (g, half, c)
    {
        const int g = t >> 2, h = (t >> 1) & 1, c = t & 1;
#pragma unroll
        for (int j = 0; j < JCOLS; ++j)
            Wbf[(size_t)(g * 2 + h) * 64 + j * 2 + c] = basis_val(j, t);
    }
    // synthesis table: t -> (ct, l); loop j -> (kg, half, c)
    {
        const int ct = t >> 4, l = t & 15;
#pragma unroll
        for (int j = 0; j < JCOLS; ++j) {
            const int kg = j >> 2, h = (j >> 1) & 1, c = j & 1;
            Wtf[(size_t)ct * 512 + kg * 64 + h * 32 + l * 2 + c] = basis_val(j, t);
        }
    }
}

__global__ void zero_f32(float* __restrict__ p, int n) {
    const int i = blockIdx.x * blockDim.x + threadIdx.x;
    if (i < n) p[i] = 0.0f;
}

// ---------------------------------------------------------------------------
// Stage 1: X(2048x32) = x(2048x8192) * Wb(8192x32), fp32 WMMA 16x16x4.
// One wave owns a 16-row block and BOTH 16-col tiles (cos+sin): a single A
// fragment feeds two WMMAs -> x is read exactly once. K split into 32 chunks
// of 256; partials merged with global f32 atomics.
// Per 4-K fragment: 1 b64 A load + 2 b64 B loads -> 2 WMMAs.
// ---------------------------------------------------------------------------
__global__ void dft_wmma(const float* __restrict__ x,
                         const float* __restrict__ Wbf,
                         float* __restrict__ X) {
    const int wave = blockIdx.x * (blockDim.x >> 5) + (threadIdx.x >> 5);
    const int lane = threadIdx.x & 31;
    const int kc   = wave & (KCHUNKS - 1);   // K chunk
    const int rt   = wave >> 5;              // row tile 0..127
    const int m0   = rt * 16;
    const int half = lane >> 4;
    const int l15  = lane & 15;

    const float* xrow = x + (size_t)(m0 + l15) * L_LEN + kc * (L_LEN / KCHUNKS) + 2 * half;
    // packed fragments for K-groups g = kc*64 .. kc*64+63
    const float* wfr  = Wbf + (size_t)kc * 8192 + half * 64 + l15 * 2;

    v8f acc0 = {};   // cos tile (cols 0..15)
    v8f acc1 = {};   // sin tile (cols 16..31)
    for (int i = 0; i < 64; ++i) {
        v2f a  = *(const v2f*)(xrow + i * 4);
        v2f b0 = *(const v2f*)(wfr + i * 128);
        v2f b1 = *(const v2f*)(wfr + i * 128 + 32);
        acc0 = __builtin_amdgcn_wmma_f32_16x16x4_f32(
            false, a, false, b0, (short)0, acc0, false, false);
        acc1 = __builtin_amdgcn_wmma_f32_16x16x4_f32(
            false, a, false, b1, (short)0, acc1, false, false);
    }
    // C layout: VGPR v -> row m0 + v + 8*half, col (lane&15) [+16 for acc1]
    float* out = X + (size_t)(m0 + 8 * half) * JCOLS + l15;
#pragma unroll
    for (int v = 0; v < 8; ++v) {
        atomicAdd(out + (size_t)v * JCOLS,      acc0[v]);
        atomicAdd(out + (size_t)v * JCOLS + 16, acc1[v]);
    }
}

// ---------------------------------------------------------------------------
// Stage 2: complex per-mode channel mixing, folded conjugate symmetry, 1/L.
//   X row r=(b*64+c): col k = Re X[k] (= sum x cos), col 16+k = sum x sin
//   (so Im X[k] = -X[r+16+k]).
//   Y[+k] = sum_c X[k]       * R[c,o,k]
//   Y[-k] = sum_c conj(X[k]) * R[c,o,25-k]
//   G cos coeff = Yr[+k] + Yr[-k] (k=0: Yr[0]); sin coeff = -Yi[+k] + Yi[-k]
// One thread per (b, o, k) with k in 0..15 (k>=13 writes zeros).
// ---------------------------------------------------------------------------
__global__ void mix_modes(const float* __restrict__ X,
                          const float* __restrict__ Wre,
                          const float* __restrict__ Wim,
                          float* __restrict__ G) {
    const int tid = blockIdx.x * blockDim.x + threadIdx.x;
    const int k = tid & 15;
    const int o = (tid >> 4) & (NC - 1);
    const int b = tid >> 10;
    if (b >= NB) return;

    float gc = 0.0f, gs = 0.0f;
    if (k <= 12) {
        float ypr = 0.f, ypi = 0.f, ymr = 0.f, ymi = 0.f;
        const int i2 = NMODES - k;   // slot of frequency -k (valid for k>=1)
        for (int c = 0; c < NC; ++c) {
            const int   r  = (b * NC + c) * JCOLS;
            const float xr = X[r + k];
            const float xs = X[r + 16 + k];
            const float xi = -xs;    // Im X[k]
            const int   wb = (c * NC + o) * NMODES;
            const float wr = Wre[wb + k];
            const float wi = Wim[wb + k];
            ypr += xr * wr - xi * wi;
            ypi += xr * wi + xi * wr;
            if (k >= 1) {
                const float wr2 = Wre[wb + i2];
                const float wi2 = Wim[wb + i2];
                // conj(X[k]) = xr + i*xs
                ymr += xr * wr2 - xs * wi2;
                ymi += xr * wi2 + xs * wr2;
            }
        }
        if (k == 0) { gc = ypr;        gs = 0.0f;      }
        else        { gc = ypr + ymr;  gs = ymi - ypi; }
    }
    const float invL = 1.0f / (float)L_LEN;
    const int   gr   = (b * NC + o) * JCOLS;
    G[gr + k]      = gc * invL;
    G[gr + 16 + k] = gs * invL;
}

// ---------------------------------------------------------------------------
// Stage 3: y(2048x8192) = G(2048x32) * Wt(32x8192), fp32 WMMA, K=32 (8 steps).
// 65536 waves; pure store-bandwidth bound (64 MB out). B fragments come from
// the packed transposed table: one contiguous b64 per lane per fragment.
// ---------------------------------------------------------------------------
__global__ void synth_wmma(const float* __restrict__ G,
                           const float* __restrict__ Wtf,
                           float* __restrict__ y) {
    const int wave = blockIdx.x * (blockDim.x >> 5) + (threadIdx.x >> 5);
    const int lane = threadIdx.x & 31;
    const int ct   = wave & 511;        // column tile (t dimension)
    const int rt   = wave >> 9;         // row tile (b*o dimension)
    const int m0   = rt * 16;
    const int t0   = ct * 16;
    const int half = lane >> 4;
    const int l15  = lane & 15;

    const float* ga = G   + (size_t)(m0 + l15) * JCOLS + 2 * half;
    const float* wp = Wtf + (size_t)ct * 512 + half * 32 + l15 * 2;

    v8f acc = {};
#pragma unroll
    for (int kg = 0; kg < 8; ++kg) {
        v2f a = *(const v2f*)(ga + kg * 4);
        v2f b = *(const v2f*)(wp + kg * 64);
        acc = __builtin_amdgcn_wmma_f32_16x16x4_f32(
            false, a, false, b, (short)0, acc, false, false);
    }
    float* orow = y + (size_t)(m0 + 8 * half) * L_LEN + (t0 + l15);
#pragma unroll
    for (int v = 0; v < 8; ++v)
        orow[(size_t)v * L_LEN] = acc[v];
}

// ---------------------------------------------------------------------------
// Launch. Workspace (floats): Wbf[262144] Wtf[262144] X[65536] G[65536]
// (= 2.5 MB). All kernels on `stream` -> ordered execution, graph-capturable.
// ---------------------------------------------------------------------------
extern "C" void kernel_launch(void* const* d_in, const int* in_sizes, int n_in,
                              void* d_out, int out_size, void* d_ws, size_t ws_size,
                              hipStream_t stream) {
    const float* x   = (const float*)d_in[0];
    const float* wre = (const float*)d_in[1];
    const float* wim = (const float*)d_in[2];
    float* out = (float*)d_out;

    float* Wbf = (float*)d_ws;                  // packed analysis basis
    float* Wtf = Wbf + (size_t)L_LEN * JCOLS;   // packed synthesis basis
    float* X   = Wtf + (size_t)JCOLS * L_LEN;   // 2048*32 atomic accumulator
    float* G   = X   + (size_t)NROWS * JCOLS;   // 2048*32

    // 1) packed basis tables (32 blocks x 256 threads)
    gen_basis<<<L_LEN / 256, 256, 0, stream>>>(Wbf, Wtf);
    // 2) zero the stage-1 atomic accumulator
    zero_f32<<<(NROWS * JCOLS) / 256, 256, 0, stream>>>(X, NROWS * JCOLS);
    // 3) analysis GEMM: 4096 waves (128 row tiles * 32 K chunks), 8 waves/block
    dft_wmma<<<(128 * KCHUNKS) / 8, 256, 0, stream>>>(x, Wbf, X);
    // 4) per-mode complex mixing -> G (32*64*16 threads)
    mix_modes<<<(NB * NC * 16) / 256, 256, 0, stream>>>(X, wre, wim, G);
    // 5) synthesis GEMM: 65536 waves (128 row tiles * 512 col tiles)
    synth_wmma<<<65536 / 8, 256, 0, stream>>>(G, Wtf, out);
}